// tempMP_52329881534842
// MI455X (gfx1250) — compile-verified
//
#include <hip/hip_runtime.h>
#include <math.h>
#include <stdint.h>

#define HGT_D   256
#define HGT_H   8
#define HGT_DK  32
#define BS_LD   40        // LDS row stride (floats) for the staged weight tile

typedef float v2f __attribute__((ext_vector_type(2)));
typedef float v8f __attribute__((ext_vector_type(8)));

// ---------------------------------------------------------------------------
// Dense GEMM: C[N x 256] = A[N x 256] @ W[256 x 256] + bias[256]
// Block = 256 threads (8 waves).  Block stages the 256x32 weight tile into LDS
// once via GLOBAL_LOAD_ASYNC_TO_LDS_B128 (ASYNCcnt), then each wave computes a
// 16x32 tile with two independent V_WMMA_F32_16X16X4_F32 accumulator chains.
// A frag (ISA 7.12.2, 32-bit A 16x4): m=lane&15, kh=lane>>4,
//   v2f = { A[m][k0+2kh], A[m][k0+2kh+1] };  B symmetric.
// C/D: vgpr i, lane l -> row = i + 8*(l>>4), col = l&15.
// ---------------------------------------------------------------------------
__global__ __launch_bounds__(256) void hgt_gemm256(const float* __restrict__ A,
                                                   const float* __restrict__ W,
                                                   const float* __restrict__ bias,
                                                   float* __restrict__ C, int N) {
    __shared__ float Bs[HGT_D * BS_LD];          // 256 rows x 40 floats = 40 KB
    const int t = threadIdx.x;
    const int colBase = blockIdx.y * 32;

    // ---- async stage W[0:256, colBase:colBase+32] -> Bs[k][c], 16B chunks ----
    {
        const unsigned ldsBase = (unsigned)(uintptr_t)Bs;
        #pragma unroll
        for (int i = 0; i < 8; ++i) {
            const int chunk = i * 256 + t;       // 2048 chunks of 16B
            const int k  = chunk >> 3;           // 8 chunks per 32-float row
            const int c4 = (chunk & 7) * 4;
            const float* gp = W + (size_t)k * HGT_D + colBase + c4;
            const unsigned lo = ldsBase + (unsigned)(k * BS_LD + c4) * 4u;
            asm volatile("global_load_async_to_lds_b128 %0, %1, off"
                         :: "v"(lo), "v"(gp) : "memory");
        }
        asm volatile("s_wait_asynccnt 0x0" ::: "memory");
    }
    __syncthreads();

    const int lane = t & 31;
    const int wave = t >> 5;
    const int m    = lane & 15;
    const int kh   = lane >> 4;                  // 0 or 1
    const int rowBase = (blockIdx.x * 8 + wave) * 16;
    if (rowBase >= N) return;                    // after barrier: safe early-out

    int rowA = rowBase + m; if (rowA >= N) rowA = N - 1;   // clamp loads
    const float* Arow = A + (size_t)rowA * HGT_D;

    v8f acc0 = {0.f, 0.f, 0.f, 0.f, 0.f, 0.f, 0.f, 0.f};
    v8f acc1 = {0.f, 0.f, 0.f, 0.f, 0.f, 0.f, 0.f, 0.f};
    #pragma unroll 4
    for (int k0 = 0; k0 < HGT_D; k0 += 4) {
        float2 av = *(const float2*)(Arow + k0 + 2 * kh);
        v2f a; a.x = av.x; a.y = av.y;
        const int r0 = (k0 + 2 * kh) * BS_LD;
        v2f b0; b0.x = Bs[r0 + m];          b0.y = Bs[r0 + BS_LD + m];
        v2f b1; b1.x = Bs[r0 + 16 + m];     b1.y = Bs[r0 + BS_LD + 16 + m];
        acc0 = __builtin_amdgcn_wmma_f32_16x16x4_f32(false, a, false, b0,
                                                     (short)0, acc0, false, false);
        acc1 = __builtin_amdgcn_wmma_f32_16x16x4_f32(false, a, false, b1,
                                                     (short)0, acc1, false, false);
    }
    const int col0 = colBase + m;
    const int col1 = colBase + 16 + m;
    const float bv0 = bias[col0];
    const float bv1 = bias[col1];
    #pragma unroll
    for (int i = 0; i < 8; ++i) {
        const int r = rowBase + i + 8 * kh;
        if (r < N) {
            C[(size_t)r * HGT_D + col0] = acc0[i] + bv0;
            C[(size_t)r * HGT_D + col1] = acc1[i] + bv1;
        }
    }
}

// ---------------------------------------------------------------------------
// Per-relation destination-q transform with pri/sqrt(DK) folded in:
//   Q'[n,h,d] = (sum_k att_w[h,d,k] * q[n,h,k]) * pri[h] / sqrt(32)
// ---------------------------------------------------------------------------
__global__ __launch_bounds__(256) void hgt_qtrans(const float* __restrict__ q,
                                                  const float* __restrict__ attw,
                                                  const float* __restrict__ pri,
                                                  float* __restrict__ Qout, int N) {
    const int gid = blockIdx.x * 256 + threadIdx.x;
    if (gid >= N * HGT_D) return;
    const int n = gid >> 8;
    const int hd = gid & 255;
    const int h = hd >> 5;
    const int d = hd & 31;
    const float* qr = q + (size_t)n * HGT_D + h * HGT_DK;
    const float* ar = attw + (size_t)(h * HGT_DK + d) * HGT_DK;
    float sum = 0.f;
    #pragma unroll
    for (int k = 0; k < HGT_DK; ++k) sum += qr[k] * ar[k];
    Qout[gid] = sum * pri[h] * 0.17677669529663687f;   // 1/sqrt(32)
}

// ---------------------------------------------------------------------------
// Edge kernel: one wave32 per edge.  Lane l covers elements [8l, 8l+8) (head
// h = l>>2).  Per-head dot via shfl_xor in 4-lane groups, w = exp(score),
// atomic scatter of w*v[src] into acc[dst] and w into ssum[dst,h].
// ---------------------------------------------------------------------------
__global__ __launch_bounds__(256) void hgt_edge(const int* __restrict__ src,
                                                const int* __restrict__ dst,
                                                const float* __restrict__ Ksrc,
                                                const float* __restrict__ Qdst,
                                                const float* __restrict__ Vsrc,
                                                float* __restrict__ acc,
                                                float* __restrict__ ssum, int nE) {
    const int lane = threadIdx.x & 31;
    const int e = blockIdx.x * 8 + (threadIdx.x >> 5);
    if (e >= nE) return;
    const int si = src[e];
    const int di = dst[e];
    const float4* kp = (const float4*)(Ksrc + (size_t)si * HGT_D + lane * 8);
    const float4* qp = (const float4*)(Qdst + (size_t)di * HGT_D + lane * 8);
    float4 k0 = kp[0], k1 = kp[1];
    float4 q0 = qp[0], q1 = qp[1];
    float p = k0.x * q0.x + k0.y * q0.y + k0.z * q0.z + k0.w * q0.w
            + k1.x * q1.x + k1.y * q1.y + k1.z * q1.z + k1.w * q1.w;
    p += __shfl_xor(p, 1, 4);
    p += __shfl_xor(p, 2, 4);
    const float w = __expf(p);                      // scores pre-scaled; shift-free softmax
    const float4* vp = (const float4*)(Vsrc + (size_t)si * HGT_D + lane * 8);
    float4 v0 = vp[0], v1 = vp[1];
    float* ap = acc + (size_t)di * HGT_D + lane * 8;
    atomicAdd(ap + 0, w * v0.x); atomicAdd(ap + 1, w * v0.y);
    atomicAdd(ap + 2, w * v0.z); atomicAdd(ap + 3, w * v0.w);
    atomicAdd(ap + 4, w * v1.x); atomicAdd(ap + 5, w * v1.y);
    atomicAdd(ap + 6, w * v1.z); atomicAdd(ap + 7, w * v1.w);
    if ((lane & 3) == 0) atomicAdd(ssum + (size_t)di * HGT_H + (lane >> 2), w);
}

// ---------------------------------------------------------------------------
// Finalize per relation: t[n,h,k] (+)= scale * relu( (acc[n,h,:]/max(s,1e-9)) @ msg_w[h] )
// ---------------------------------------------------------------------------
__global__ __launch_bounds__(256) void hgt_msg_relu(const float* __restrict__ acc,
                                                    const float* __restrict__ s,
                                                    const float* __restrict__ msgw,
                                                    float* __restrict__ t, int N,
                                                    float scale, int accum) {
    const int gid = blockIdx.x * 256 + threadIdx.x;
    if (gid >= N * HGT_D) return;
    const int n = gid >> 8;
    const int hk = gid & 255;
    const int h = hk >> 5;
    const int k = hk & 31;
    float denom = fmaxf(s[(size_t)n * HGT_H + h], 1e-9f);
    const float* ar = acc + (size_t)n * HGT_D + h * HGT_DK;
    const float* mw = msgw + (size_t)h * HGT_DK * HGT_DK + k;   // [h][dd][k], dd stride 32
    float sum = 0.f;
    #pragma unroll
    for (int dd = 0; dd < HGT_DK; ++dd) sum += ar[dd] * mw[dd * HGT_DK];
    float val = fmaxf(sum / denom, 0.f) * scale;
    if (accum) t[gid] += val; else t[gid] = val;
}

// ---------------------------------------------------------------------------
// Gated residual + LayerNorm.  One block (256 threads) per node.
// ---------------------------------------------------------------------------
__global__ __launch_bounds__(256) void hgt_gate_ln(const float* __restrict__ trans,
                                                   const float* __restrict__ feat,
                                                   const float* __restrict__ skip, int skipIdx,
                                                   const float* __restrict__ gamma,
                                                   const float* __restrict__ beta,
                                                   float* __restrict__ out) {
    __shared__ float red[256];
    const int n = blockIdx.x;
    const int t = threadIdx.x;
    const float a = 1.f / (1.f + __expf(-skip[skipIdx]));
    const float x = trans[(size_t)n * HGT_D + t] * a + feat[(size_t)n * HGT_D + t] * (1.f - a);
    red[t] = x; __syncthreads();
    for (int o = 128; o > 0; o >>= 1) { if (t < o) red[t] += red[t + o]; __syncthreads(); }
    const float mu = red[0] * (1.f / HGT_D);
    __syncthreads();
    const float d = x - mu;
    red[t] = d * d; __syncthreads();
    for (int o = 128; o > 0; o >>= 1) { if (t < o) red[t] += red[t + o]; __syncthreads(); }
    const float var = red[0] * (1.f / HGT_D);
    out[(size_t)n * HGT_D + t] = d * rsqrtf(var + 1e-5f) * gamma[t] + beta[t];
}

// ---------------------------------------------------------------------------
extern "C" void kernel_launch(void* const* d_in, const int* in_sizes, int n_in,
                              void* d_out, int out_size, void* d_ws, size_t ws_size,
                              hipStream_t stream) {
    const float* feat_word  = (const float*)d_in[0];
    const float* feat_topic = (const float*)d_in[1];
    const float* feat_doc   = (const float*)d_in[2];
    const float* Wk = (const float*)d_in[3];
    const float* bk = (const float*)d_in[4];
    const float* Wq = (const float*)d_in[5];
    const float* bq = (const float*)d_in[6];
    const float* Wv = (const float*)d_in[7];
    const float* bv = (const float*)d_in[8];
    const float* Wa = (const float*)d_in[9];
    const float* ba = (const float*)d_in[10];
    const float* ln_scale = (const float*)d_in[11];
    const float* ln_bias  = (const float*)d_in[12];
    const float* skip     = (const float*)d_in[13];
    const float* rel_pri  = (const float*)d_in[14];
    const float* rel_att  = (const float*)d_in[15];
    const float* rel_msg  = (const float*)d_in[16];

    const int Nw = in_sizes[0] / HGT_D;
    const int Nt = in_sizes[1] / HGT_D;
    const int Nd = in_sizes[2] / HGT_D;

    const int* src_e[5]; const int* dst_e[5]; int nE[5];
    for (int r = 0; r < 5; ++r) {
        src_e[r] = (const int*)d_in[17 + 2 * r];
        dst_e[r] = (const int*)d_in[18 + 2 * r];
        nE[r]    = in_sizes[17 + 2 * r];
    }

    float* out = (float*)d_out;
    float* ws  = (float*)d_ws;

    // ---- workspace layout (floats) ----
    size_t o = 0;
    float* k_w = ws + o; o += (size_t)Nw * HGT_D;
    float* q_w = ws + o; o += (size_t)Nw * HGT_D;
    float* v_w = ws + o; o += (size_t)Nw * HGT_D;
    float* k_t = ws + o; o += (size_t)Nt * HGT_D;
    float* q_t = ws + o; o += (size_t)Nt * HGT_D;
    float* v_t = ws + o; o += (size_t)Nt * HGT_D;
    float* q_d = ws + o; o += (size_t)Nd * HGT_D;
    float* Qbuf   = ws + o; o += (size_t)Nw * HGT_D;   // max n_dst = Nw
    float* accbuf = ws + o; o += (size_t)Nw * HGT_D;
    float* sbuf   = ws + o; o += (size_t)Nw * HGT_H;
    float* t_word  = ws + o; o += (size_t)Nw * HGT_D;
    float* t_topic = ws + o; o += (size_t)Nt * HGT_D;
    float* t_doc   = ws + o; o += (size_t)Nd * HGT_D;
    // trans buffers alias projections (dead after the edge phase)
    float* trans_word  = k_w;
    float* trans_topic = k_t;
    float* trans_doc   = q_d;

    const size_t WOFF = (size_t)HGT_D * HGT_D;   // 65536 floats per ntype weight
    dim3 blk(256);
    const int CT = HGT_D / 32;                   // 8 column tiles of 32

    // ---- phase 1: projections (WMMA GEMMs) ----
    {
        dim3 gw((Nw + 127) / 128, CT);
        dim3 gt((Nt + 127) / 128, CT);
        dim3 gd((Nd + 127) / 128, CT);
        hipLaunchKernelGGL(hgt_gemm256, gw, blk, 0, stream, feat_word,  Wk,            bk,            k_w, Nw);
        hipLaunchKernelGGL(hgt_gemm256, gw, blk, 0, stream, feat_word,  Wq,            bq,            q_w, Nw);
        hipLaunchKernelGGL(hgt_gemm256, gw, blk, 0, stream, feat_word,  Wv,            bv,            v_w, Nw);
        hipLaunchKernelGGL(hgt_gemm256, gt, blk, 0, stream, feat_topic, Wk + WOFF,     bk + HGT_D,    k_t, Nt);
        hipLaunchKernelGGL(hgt_gemm256, gt, blk, 0, stream, feat_topic, Wq + WOFF,     bq + HGT_D,    q_t, Nt);
        hipLaunchKernelGGL(hgt_gemm256, gt, blk, 0, stream, feat_topic, Wv + WOFF,     bv + HGT_D,    v_t, Nt);
        hipLaunchKernelGGL(hgt_gemm256, gd, blk, 0, stream, feat_doc,   Wq + 2 * WOFF, bq + 2*HGT_D,  q_d, Nd);
    }

    // ---- phase 2: per-relation attention + aggregation (relation-sequential) ----
    struct RelCfg {
        const float* Ks; const float* Vs; const float* q;
        int ndst; float* t; float scale; int accum;
    };
    RelCfg rels[5] = {
        { k_w, v_w, q_w, Nw, t_word,  1.0f, 0 },   // ww
        { k_w, v_w, q_t, Nt, t_topic, 0.5f, 0 },   // wt
        { k_w, v_w, q_d, Nd, t_doc,   0.5f, 0 },   // wd
        { k_t, v_t, q_t, Nt, t_topic, 0.5f, 1 },   // tt
        { k_t, v_t, q_d, Nd, t_doc,   0.5f, 1 },   // td
    };
    const size_t RELW = (size_t)HGT_H * HGT_DK * HGT_DK;   // 8192 floats per relation
    for (int r = 0; r < 5; ++r) {
        const RelCfg& rc = rels[r];
        hipLaunchKernelGGL(hgt_qtrans, dim3(rc.ndst), blk, 0, stream,
                           rc.q, rel_att + r * RELW, rel_pri + r * HGT_H, Qbuf, rc.ndst);
        hipMemsetAsync(accbuf, 0, (size_t)rc.ndst * HGT_D * sizeof(float), stream);
        hipMemsetAsync(sbuf,   0, (size_t)rc.ndst * HGT_H * sizeof(float), stream);
        hipLaunchKernelGGL(hgt_edge, dim3((nE[r] + 7) / 8), blk, 0, stream,
                           src_e[r], dst_e[r], rc.Ks, Qbuf, rc.Vs, accbuf, sbuf, nE[r]);
        hipLaunchKernelGGL(hgt_msg_relu, dim3(rc.ndst), blk, 0, stream,
                           accbuf, sbuf, rel_msg + r * RELW, rc.t, rc.ndst, rc.scale, rc.accum);
    }

    // ---- phase 3: output transform (WMMA GEMMs) + gated residual + LayerNorm ----
    {
        dim3 gw((Nw + 127) / 128, CT);
        dim3 gt((Nt + 127) / 128, CT);
        dim3 gd((Nd + 127) / 128, CT);
        hipLaunchKernelGGL(hgt_gemm256, gw, blk, 0, stream, t_word,  Wa,            ba,           trans_word,  Nw);
        hipLaunchKernelGGL(hgt_gemm256, gt, blk, 0, stream, t_topic, Wa + WOFF,     ba + HGT_D,   trans_topic, Nt);
        hipLaunchKernelGGL(hgt_gemm256, gd, blk, 0, stream, t_doc,   Wa + 2 * WOFF, ba + 2*HGT_D, trans_doc,   Nd);

        hipLaunchKernelGGL(hgt_gate_ln, dim3(Nw), blk, 0, stream,
                           trans_word,  feat_word,  skip, 0, ln_scale,              ln_bias,              out);
        hipLaunchKernelGGL(hgt_gate_ln, dim3(Nt), blk, 0, stream,
                           trans_topic, feat_topic, skip, 1, ln_scale + HGT_D,      ln_bias + HGT_D,      out + (size_t)Nw * HGT_D);
        hipLaunchKernelGGL(hgt_gate_ln, dim3(Nd), blk, 0, stream,
                           trans_doc,   feat_doc,   skip, 2, ln_scale + 2 * HGT_D,  ln_bias + 2 * HGT_D,  out + (size_t)(Nw + Nt) * HGT_D);
    }
}